// Similarity_41420664602989
// MI455X (gfx1250) — compile-verified
//
#include <hip/hip_runtime.h>
#include <math.h>

#define NROWS  4096
#define DIM    1024
#define TOPK   10
#define CEPS   1e-8f
#define WEIGHT 1.0f
#define NREG   (NROWS / 32)                 // 128 regions per dim
#define NTRI   (NREG * (NREG + 1) / 2)      // 8256 upper-triangular regions

typedef float v2f __attribute__((ext_vector_type(2)));
typedef float v8f __attribute__((ext_vector_type(8)));

// ---------------------------------------------------------------------------
// Kernel 1: row L2 norms for both matrices. One wave32 per row.
// ---------------------------------------------------------------------------
__global__ __launch_bounds__(256) void norms_kernel(const float* __restrict__ xo,
                                                    const float* __restrict__ xt,
                                                    float* __restrict__ norms) {
  const int lane = threadIdx.x & 31;
  const int wave = threadIdx.x >> 5;
  const int row  = blockIdx.x * 8 + wave;
  const float* src = (blockIdx.y == 0) ? xo : xt;
  const float4* r4 = (const float4*)(src + (size_t)row * DIM);
  float s = 0.f;
#pragma unroll
  for (int i = 0; i < DIM / 4 / 32; ++i) {  // 8 x float4 per lane
    float4 v = r4[lane + i * 32];
    s += v.x * v.x + v.y * v.y + v.z * v.z + v.w * v.w;
  }
#pragma unroll
  for (int off = 16; off > 0; off >>= 1) s += __shfl_xor(s, off, 32);
  if (lane == 0) norms[blockIdx.y * NROWS + row] = sqrtf(s);
}

// ---------------------------------------------------------------------------
// Kernel 2: fused dual Gram + cosine + squared-error, upper-triangular only.
// err is symmetric (both Grams are), so each wave computes one 32x32 region
// with ry <= rx (2x2 WMMA f32 16x16x4 tiles, both inputs) and stores the
// result to both the (i,j) patch and the mirrored (j,i) patch. This halves
// the WMMA work of the dominant, compute-bound kernel.
//
// f32 16x16x4 fragment layout (ISA 7.12.2):
//   A (16x4):  lane L holds row M=L&15, K = {2*(L>>4), 2*(L>>4)+1}  -> v2f
//   B (4x16):  lane L holds col N=L&15, K = {2*(L>>4), 2*(L>>4)+1}  -> v2f
//   C/D:       VGPR r, lane L -> element (M = r + 8*(L>>4), N = L&15)
// ---------------------------------------------------------------------------
__device__ __forceinline__ int tri_offset(int r) {
  return NREG * r - (r * (r - 1)) / 2;      // regions before triangular row r
}

__global__ __launch_bounds__(256) void gram_err_kernel(const float* __restrict__ xo,
                                                       const float* __restrict__ xt,
                                                       const float* __restrict__ norms,
                                                       float* __restrict__ err) {
  const int lane = threadIdx.x;          // 0..31
  const int l = lane & 15;
  const int h = lane >> 4;

  // Decode linear wave index -> upper-triangular region pair (ry <= rx).
  const int w = blockIdx.x * 8 + threadIdx.y;     // 0 .. NTRI-1
  int ry = (int)floorf((2.0f * NREG + 1.0f -
                        sqrtf((2.0f * NREG + 1.0f) * (2.0f * NREG + 1.0f) - 8.0f * (float)w)) * 0.5f);
  if (ry > NREG - 1) ry = NREG - 1;
  if (ry < 0) ry = 0;
  while (tri_offset(ry) > w) --ry;
  while (ry < NREG - 1 && tri_offset(ry + 1) <= w) ++ry;
  const int rx = ry + (w - tri_offset(ry));

  const int i0 = ry * 32;
  const int j0 = rx * 32;

  v8f co00 = {}, co01 = {}, co10 = {}, co11 = {};
  v8f ct00 = {}, ct01 = {}, ct10 = {}, ct11 = {};

  // ---- Gram of model_output ----
  {
    const float* pa0 = xo + (size_t)(i0 + l) * DIM + 2 * h;
    const float* pa1 = xo + (size_t)(i0 + 16 + l) * DIM + 2 * h;
    const float* pb0 = xo + (size_t)(j0 + l) * DIM + 2 * h;
    const float* pb1 = xo + (size_t)(j0 + 16 + l) * DIM + 2 * h;
#pragma unroll 4
    for (int k = 0; k < DIM; k += 4) {
      v2f a0 = *(const v2f*)(pa0 + k);
      v2f a1 = *(const v2f*)(pa1 + k);
      v2f b0 = *(const v2f*)(pb0 + k);
      v2f b1 = *(const v2f*)(pb1 + k);
      co00 = __builtin_amdgcn_wmma_f32_16x16x4_f32(false, a0, false, b0, (short)0, co00, false, false);
      co01 = __builtin_amdgcn_wmma_f32_16x16x4_f32(false, a0, false, b1, (short)0, co01, false, false);
      co10 = __builtin_amdgcn_wmma_f32_16x16x4_f32(false, a1, false, b0, (short)0, co10, false, false);
      co11 = __builtin_amdgcn_wmma_f32_16x16x4_f32(false, a1, false, b1, (short)0, co11, false, false);
    }
  }
  // ---- Gram of targets ----
  {
    const float* pa0 = xt + (size_t)(i0 + l) * DIM + 2 * h;
    const float* pa1 = xt + (size_t)(i0 + 16 + l) * DIM + 2 * h;
    const float* pb0 = xt + (size_t)(j0 + l) * DIM + 2 * h;
    const float* pb1 = xt + (size_t)(j0 + 16 + l) * DIM + 2 * h;
#pragma unroll 4
    for (int k = 0; k < DIM; k += 4) {
      v2f a0 = *(const v2f*)(pa0 + k);
      v2f a1 = *(const v2f*)(pa1 + k);
      v2f b0 = *(const v2f*)(pb0 + k);
      v2f b1 = *(const v2f*)(pb1 + k);
      ct00 = __builtin_amdgcn_wmma_f32_16x16x4_f32(false, a0, false, b0, (short)0, ct00, false, false);
      ct01 = __builtin_amdgcn_wmma_f32_16x16x4_f32(false, a0, false, b1, (short)0, ct01, false, false);
      ct10 = __builtin_amdgcn_wmma_f32_16x16x4_f32(false, a1, false, b0, (short)0, ct10, false, false);
      ct11 = __builtin_amdgcn_wmma_f32_16x16x4_f32(false, a1, false, b1, (short)0, ct11, false, false);
    }
  }

  // ---- cosine normalize, zero diagonal, squared error, mirrored store ----
  const float* no = norms;
  const float* nt = norms + NROWS;
#pragma unroll
  for (int ti = 0; ti < 2; ++ti) {
#pragma unroll
    for (int r = 0; r < 8; ++r) {
      const int gi = i0 + ti * 16 + r + 8 * h;
      const float noi = no[gi];
      const float nti = nt[gi];
#pragma unroll
      for (int tj = 0; tj < 2; ++tj) {
        const int gj = j0 + tj * 16 + l;
        const v8f& vco = ti ? (tj ? co11 : co10) : (tj ? co01 : co00);
        const v8f& vct = ti ? (tj ? ct11 : ct10) : (tj ? ct01 : ct00);
        const float c1 = vco[r] / fmaxf(noi * no[gj], CEPS);
        const float c2 = vct[r] / fmaxf(nti * nt[gj], CEPS);
        const float d = c1 - c2;
        const float e = (gi == gj) ? 0.f : d * d;
        err[(size_t)gi * NROWS + gj] = e;   // upper patch (coalesced)
        err[(size_t)gj * NROWS + gi] = e;   // mirrored lower patch
      }
    }
  }
}

// ---------------------------------------------------------------------------
// Kernel 3: per-row top-10 sum. Block (256) per row; register insertion sort
// + deterministic LDS pairwise merge of sorted 10-lists.
// ---------------------------------------------------------------------------
__device__ __forceinline__ void insert10(float (&t)[TOPK], float v) {
  if (v <= t[TOPK - 1]) return;
  t[TOPK - 1] = v;
#pragma unroll
  for (int i = TOPK - 1; i > 0; --i) {
    if (t[i] > t[i - 1]) { float tmp = t[i]; t[i] = t[i - 1]; t[i - 1] = tmp; }
  }
}

__global__ __launch_bounds__(256) void topk_kernel(const float* __restrict__ err,
                                                   float* __restrict__ rowsum) {
  __shared__ float lds[256 * TOPK];
  const int row = blockIdx.x;
  const int t = threadIdx.x;
  const float* r = err + (size_t)row * NROWS;

  float top[TOPK];
#pragma unroll
  for (int i = 0; i < TOPK; ++i) top[i] = -1.f;
#pragma unroll
  for (int i = 0; i < NROWS / 256; ++i) insert10(top, r[t + i * 256]);

#pragma unroll
  for (int i = 0; i < TOPK; ++i) lds[t * TOPK + i] = top[i];
  __syncthreads();

  for (int stride = 128; stride > 0; stride >>= 1) {
    if (t < stride) {
      float a[TOPK], b[TOPK], o[TOPK];
#pragma unroll
      for (int i = 0; i < TOPK; ++i) { a[i] = lds[t * TOPK + i]; b[i] = lds[(t + stride) * TOPK + i]; }
      int ia = 0, ib = 0;
#pragma unroll
      for (int i = 0; i < TOPK; ++i) o[i] = (a[ia] >= b[ib]) ? a[ia++] : b[ib++];
#pragma unroll
      for (int i = 0; i < TOPK; ++i) lds[t * TOPK + i] = o[i];
    }
    __syncthreads();
  }

  if (t == 0) {
    float s = 0.f;
#pragma unroll
    for (int i = 0; i < TOPK; ++i) s += lds[i];
    rowsum[row] = s;
  }
}

// ---------------------------------------------------------------------------
// Kernel 4: deterministic final reduction -> scalar loss.
// ---------------------------------------------------------------------------
__global__ __launch_bounds__(256) void finalize_kernel(const float* __restrict__ rowsum,
                                                       float* __restrict__ out) {
  __shared__ float lds[256];
  const int t = threadIdx.x;
  float s = 0.f;
#pragma unroll
  for (int i = 0; i < NROWS / 256; ++i) s += rowsum[t + i * 256];
  lds[t] = s;
  __syncthreads();
  for (int stride = 128; stride > 0; stride >>= 1) {
    if (t < stride) lds[t] += lds[t + stride];
    __syncthreads();
  }
  if (t == 0) out[0] = WEIGHT * lds[0] / (float)(NROWS * TOPK);
}

// ---------------------------------------------------------------------------
extern "C" void kernel_launch(void* const* d_in, const int* in_sizes, int n_in,
                              void* d_out, int out_size, void* d_ws, size_t ws_size,
                              hipStream_t stream) {
  const float* xo = (const float*)d_in[0];   // model_output [4096,1024] f32
  const float* xt = (const float*)d_in[1];   // targets      [4096,1024] f32

  float* norms  = (float*)d_ws;                          // 2*4096 floats
  float* rowsum = norms + 2 * NROWS;                     // 4096 floats
  float* err    = (float*)((char*)d_ws + (1 << 16));     // 4096*4096 floats (64 MB)

  norms_kernel<<<dim3(NROWS / 8, 2), 256, 0, stream>>>(xo, xt, norms);
  gram_err_kernel<<<dim3(NTRI / 8), dim3(32, 8), 0, stream>>>(xo, xt, norms, err);
  topk_kernel<<<NROWS, 256, 0, stream>>>(err, rowsum);
  finalize_kernel<<<1, 256, 0, stream>>>(rowsum, (float*)d_out);
}